// StructureExtractor_9328668967145
// MI455X (gfx1250) — compile-verified
//
#include <hip/hip_runtime.h>

// ---------------- CDNA5 WMMA (fp32, 16x16x4) ----------------
typedef __attribute__((ext_vector_type(2))) float v2f;
typedef __attribute__((ext_vector_type(8))) float v8f;

__device__ __forceinline__ v8f wmma4(v2f a, v2f b, v8f c) {
  // D = A(16x4) * B(4x16) + C(16x16), fp32 throughout
  return __builtin_amdgcn_wmma_f32_16x16x4_f32(false, a, false, b, (short)0, c,
                                               false, false);
}

// ---------------- CDNA5 async memory->LDS (ASYNCcnt path) ----------------
typedef __attribute__((address_space(3))) void lds_void_t;

__device__ __forceinline__ unsigned lds_offset_of(const void* p) {
  // generic -> LDS addrspace cast, then ptrtoint gives the 32-bit LDS offset
  return (unsigned)(unsigned long long)(lds_void_t*)p;
}

__device__ __forceinline__ void async_gather_b128(unsigned lds_off, const float* gsrc) {
  unsigned long long ga = (unsigned long long)(uintptr_t)gsrc;
  // per-lane: copies 16 bytes from the global address in ga to the LDS offset in lds_off; tracked with ASYNCcnt
  asm volatile("global_load_async_to_lds_b128 %0, %1, off"
               :: "v"(lds_off), "v"(ga)
               : "memory");
}

__device__ __forceinline__ void wait_async0() {
  asm volatile("s_wait_asynccnt 0x0" ::: "memory");
}

// ---------------- model hyperparameters ----------------
constexpr int Nn    = 20000;
constexpr int Ee    = 320000;
constexpr int Dd    = 128;
constexpr int Tt    = 8;
constexpr int Ff    = 16;
constexpr int Ll    = 3;
constexpr int Ss    = 4096;
constexpr int INNER = 512;      // (L+1)*D
constexpr float BN_EPS = 1e-5f;

// padded LDS row strides (stride mod 64 == 4 -> conflict-free A reads;
// stride*4 bytes is a multiple of 16 -> b128 async destinations stay aligned)
constexpr int EPAD = 260;  // for 256-wide cat tiles   (1040 B rows)
constexpr int MPAD = 132;  // for 128-wide mid tile    (528 B rows)
constexpr int BPAD = 516;  // for 512-wide pooled tile (2064 B rows)

// ---------------- small utility kernels ----------------
__global__ void fill_zero_kernel(float* __restrict__ p, int n) {
  int i = blockIdx.x * blockDim.x + threadIdx.x;
  if (i < n) p[i] = 0.0f;
}

__global__ void init_xc_kernel(const float* __restrict__ x, float* __restrict__ xc) {
  int i = blockIdx.x * blockDim.x + threadIdx.x;
  if (i >= Nn * Dd) return;
  int node = i >> 7, col = i & 127;
  xc[(size_t)node * INNER + col] = x[i];
}

// segment-sum of xc rows into pooled (8.4 MB, L2-resident atomics)
__global__ void pool_kernel(const float* __restrict__ xc, const int* __restrict__ sg,
                            float* __restrict__ pooled) {
  int i = blockIdx.x * blockDim.x + threadIdx.x;
  if (i >= Nn * INNER) return;
  int node = i >> 9, col = i & 511;
  atomicAdd(pooled + (size_t)sg[node] * INNER + col, xc[i]);
}

// ---------------- edge message kernel (per layer) ----------------
// one wave = 16 edges; gather 32 rows via async global->LDS (one in-flight burst),
// per tower: [16x32] @ [32x16] via 8 chained WMMAs, then fp32 atomic scatter-add
// into agg[dst] (10 MB, L2-resident).
__global__ void __launch_bounds__(128)
edge_msg_kernel(const float* __restrict__ hin,   // xc + l*D, row stride INNER
                const int* __restrict__ srcE, const int* __restrict__ dstE,
                const float* __restrict__ pw,    // [T][F][2F]
                const float* __restrict__ pb,    // [T][F]
                float* __restrict__ agg)         // [N][128]
{
  extern __shared__ float smem[];
  const int lane = threadIdx.x & 31;
  const int wave = threadIdx.x >> 5;
  const int tile = blockIdx.x * 4 + wave;       // uniform per wave -> EXEC stays full
  if (tile >= Ee / 16) return;
  float* mt = smem + wave * (16 * EPAD);
  const unsigned mt_off = lds_offset_of(mt);

  const int e0 = tile * 16;
  // gather: cols [0,128) = h[dst], cols [128,256) = h[src]; 32 async b128 bursts
  for (int r = 0; r < 32; ++r) {
    const int e = e0 + (r & 15);
    const int node = (r < 16) ? dstE[e] : srcE[e];
    const float* gsrc = hin + (size_t)node * INNER + lane * 4;
    const unsigned loff =
        mt_off + (unsigned)(((r & 15) * EPAD + ((r < 16) ? 0 : 128) + lane * 4) * 4);
    async_gather_b128(loff, gsrc);
  }
  wait_async0();

  const int mn = lane & 15;       // A-row (edge) == B-col (out feature)
  const int khalf = lane >> 4;    // K sub-half per ISA fragment layout
  int dn[8];
#pragma unroll
  for (int i = 0; i < 8; ++i) dn[i] = dstE[e0 + khalf * 8 + i];

  const float* arow = mt + mn * EPAD;
  for (int t = 0; t < Tt; ++t) {
    const float bias = pb[t * Ff + mn];
    v8f acc;
#pragma unroll
    for (int i = 0; i < 8; ++i) acc[i] = bias;
    const float* wrow = pw + (t * Ff + mn) * (2 * Ff);
#pragma unroll
    for (int kk = 0; kk < 8; ++kk) {
      const int kb = kk * 4 + khalf * 2;                       // even: pairs never straddle 16
      const int col = (kb < 16) ? (t * 16 + kb) : (112 + t * 16 + kb);
      const float2 av = *(const float2*)(arow + col);
      const float2 bv = *(const float2*)(wrow + kb);           // B[k][g] = w[g][k]
      v2f a; a.x = av.x; a.y = av.y;
      v2f b; b.x = bv.x; b.y = bv.y;
      acc = wmma4(a, b, acc);
    }
#pragma unroll
    for (int i = 0; i < 8; ++i)
      atomicAdd(agg + (size_t)dn[i] * Dd + t * Ff + mn, acc[i]);
  }
}

// ---------------- node update kernel (per layer) ----------------
// one wave = 16 nodes: async-stage cat(x,agg) tile, per-tower post linear (WMMA),
// then 128x128 lin (WMMA), ReLU, write into xc column block (layer+1)*D.
__global__ void __launch_bounds__(128)
node_update_kernel(float* __restrict__ xc, const int layer,
                   const float* __restrict__ agg,
                   const float* __restrict__ pw, const float* __restrict__ pb,
                   const float* __restrict__ lw, const float* __restrict__ lb)
{
  extern __shared__ float smem[];
  const int lane = threadIdx.x & 31;
  const int wave = threadIdx.x >> 5;
  const int tile = blockIdx.x * 4 + wave;
  if (tile >= Nn / 16) return;
  float* in_t = smem + wave * (16 * EPAD + 16 * MPAD);
  float* mid  = in_t + 16 * EPAD;
  const unsigned in_off = lds_offset_of(in_t);

  const int n0 = tile * 16;
  const float* hin = xc + layer * Dd;
  for (int r = 0; r < 32; ++r) {
    const int node = n0 + (r & 15);
    const float* gsrc = (r < 16)
        ? (hin + (size_t)node * INNER + lane * 4)
        : (agg + (size_t)node * Dd + lane * 4);
    const unsigned loff =
        in_off + (unsigned)(((r & 15) * EPAD + ((r < 16) ? 0 : 128) + lane * 4) * 4);
    async_gather_b128(loff, gsrc);
  }
  wait_async0();

  const int mn = lane & 15;
  const int khalf = lane >> 4;
  const float* arow = in_t + mn * EPAD;

  // per-tower post linear -> mid[16][128] in LDS (same-wave DS ops are in-order)
  for (int t = 0; t < Tt; ++t) {
    const float bias = pb[t * Ff + mn];
    v8f acc;
#pragma unroll
    for (int i = 0; i < 8; ++i) acc[i] = bias;
    const float* wrow = pw + (t * Ff + mn) * (2 * Ff);
#pragma unroll
    for (int kk = 0; kk < 8; ++kk) {
      const int kb = kk * 4 + khalf * 2;
      const int col = (kb < 16) ? (t * 16 + kb) : (112 + t * 16 + kb);
      const float2 av = *(const float2*)(arow + col);
      const float2 bv = *(const float2*)(wrow + kb);
      v2f a; a.x = av.x; a.y = av.y;
      v2f b; b.x = bv.x; b.y = bv.y;
      acc = wmma4(a, b, acc);
    }
#pragma unroll
    for (int i = 0; i < 8; ++i)
      mid[(khalf * 8 + i) * MPAD + t * Ff + mn] = acc[i];
  }

  // lin: [16x128] @ lin_w^T [128x128] + bias, ReLU, store
  const float* amid = mid + mn * MPAD;
  for (int nt = 0; nt < 8; ++nt) {
    const int j0 = nt * 16;
    const float bias = lb[j0 + mn];
    v8f acc;
#pragma unroll
    for (int i = 0; i < 8; ++i) acc[i] = bias;
    const float* wcol = lw + (size_t)(j0 + mn) * Dd;           // B[k][j] = lin_w[j][k]
#pragma unroll
    for (int kk = 0; kk < 32; ++kk) {
      const int kb = kk * 4 + khalf * 2;
      const float2 av = *(const float2*)(amid + kb);
      const float2 bv = *(const float2*)(wcol + kb);
      v2f a; a.x = av.x; a.y = av.y;
      v2f b; b.x = bv.x; b.y = bv.y;
      acc = wmma4(a, b, acc);
    }
#pragma unroll
    for (int i = 0; i < 8; ++i) {
      const int node = n0 + khalf * 8 + i;
      xc[(size_t)node * INNER + (layer + 1) * Dd + j0 + mn] = fmaxf(acc[i], 0.0f);
    }
  }
}

// ---------------- BN + output projection ----------------
// one wave = 16 subgraph rows; BN applied on LDS fill (must pass through VGPRs,
// so direct loads here); [16x512] @ op_w^T [512x128].
__global__ void __launch_bounds__(128)
bn_outproj_kernel(const float* __restrict__ pooled,
                  const float* __restrict__ gamma, const float* __restrict__ beta,
                  const float* __restrict__ mean,  const float* __restrict__ var,
                  const float* __restrict__ opw,   const float* __restrict__ opb,
                  float* __restrict__ out)
{
  extern __shared__ float smem[];
  const int lane = threadIdx.x & 31;
  const int wave = threadIdx.x >> 5;
  const int tile = blockIdx.x * 4 + wave;
  if (tile >= Ss / 16) return;
  float* xt = smem + wave * (16 * BPAD);

  const int s0 = tile * 16;
  for (int r = 0; r < 16; ++r) {
    const float* prow = pooled + (size_t)(s0 + r) * INNER;
#pragma unroll
    for (int c4 = 0; c4 < 4; ++c4) {
      const int col = c4 * 128 + lane * 4;
      const float4 p = *(const float4*)(prow + col);
      float o[4] = {p.x, p.y, p.z, p.w};
#pragma unroll
      for (int q = 0; q < 4; ++q) {
        const int c = col + q;
        o[q] = (o[q] - mean[c]) * rsqrtf(var[c] + BN_EPS) * gamma[c] + beta[c];
      }
      float* dstp = xt + r * BPAD + col;
      dstp[0] = o[0]; dstp[1] = o[1]; dstp[2] = o[2]; dstp[3] = o[3];
    }
  }

  const int mn = lane & 15;
  const int khalf = lane >> 4;
  const float* arow = xt + mn * BPAD;
  for (int nt = 0; nt < 8; ++nt) {
    const int j0 = nt * 16;
    const float bias = opb[j0 + mn];
    v8f acc;
#pragma unroll
    for (int i = 0; i < 8; ++i) acc[i] = bias;
    const float* wcol = opw + (size_t)(j0 + mn) * INNER;       // B[k][j] = op_w[j][k]
    for (int kk = 0; kk < 128; ++kk) {
      const int kb = kk * 4 + khalf * 2;
      const float2 av = *(const float2*)(arow + kb);
      const float2 bv = *(const float2*)(wcol + kb);
      v2f a; a.x = av.x; a.y = av.y;
      v2f b; b.x = bv.x; b.y = bv.y;
      acc = wmma4(a, b, acc);
    }
#pragma unroll
    for (int i = 0; i < 8; ++i)
      out[(size_t)(s0 + khalf * 8 + i) * Dd + j0 + mn] = acc[i];
  }
}

// ---------------- launch ----------------
extern "C" void kernel_launch(void* const* d_in, const int* in_sizes, int n_in,
                              void* d_out, int out_size, void* d_ws, size_t ws_size,
                              hipStream_t stream) {
  (void)in_sizes; (void)n_in; (void)out_size; (void)ws_size;
  const float* x        = (const float*)d_in[0];
  const int*   ei       = (const int*)d_in[1];   // [2][E]
  const int*   sg       = (const int*)d_in[2];   // [N]
  const float* pre_w    = (const float*)d_in[3]; // [L][T][F][2F]
  const float* pre_b    = (const float*)d_in[4]; // [L][T][F]
  const float* post_w   = (const float*)d_in[5];
  const float* post_b   = (const float*)d_in[6];
  const float* lin_w    = (const float*)d_in[7]; // [L][D][D]
  const float* lin_b    = (const float*)d_in[8]; // [L][D]
  const float* bn_gamma = (const float*)d_in[9];
  const float* bn_beta  = (const float*)d_in[10];
  const float* bn_mean  = (const float*)d_in[11];
  const float* bn_var   = (const float*)d_in[12];
  const float* op_w     = (const float*)d_in[13]; // [D][INNER]
  const float* op_b     = (const float*)d_in[14]; // [D]
  float* out = (float*)d_out;

  char* ws = (char*)d_ws;
  float* xc     = (float*)ws;                                       // N*512
  float* agg    = (float*)(ws + (size_t)Nn * INNER * 4);            // N*128
  float* pooled = (float*)(ws + (size_t)Nn * INNER * 4
                              + (size_t)Nn * Dd * 4);               // S*512

  const int* srcE = ei;        // edge_index[0]
  const int* dstE = ei + Ee;   // edge_index[1]

  const size_t edge_lds = 4 * 16 * EPAD * sizeof(float);
  const size_t node_lds = 4 * (16 * EPAD + 16 * MPAD) * sizeof(float);
  const size_t bn_lds   = 4 * 16 * BPAD * sizeof(float);

  init_xc_kernel<<<(Nn * Dd + 255) / 256, 256, 0, stream>>>(x, xc);

  for (int l = 0; l < Ll; ++l) {
    fill_zero_kernel<<<(Nn * Dd + 255) / 256, 256, 0, stream>>>(agg, Nn * Dd);
    edge_msg_kernel<<<Ee / 16 / 4, 128, edge_lds, stream>>>(
        xc + l * Dd, srcE, dstE,
        pre_w + (size_t)l * Tt * Ff * 2 * Ff, pre_b + (size_t)l * Tt * Ff, agg);
    node_update_kernel<<<(Nn / 16 + 3) / 4, 128, node_lds, stream>>>(
        xc, l, agg,
        post_w + (size_t)l * Tt * Ff * 2 * Ff, post_b + (size_t)l * Tt * Ff,
        lin_w + (size_t)l * Dd * Dd, lin_b + (size_t)l * Dd);
  }

  fill_zero_kernel<<<(Ss * INNER + 255) / 256, 256, 0, stream>>>(pooled, Ss * INNER);
  pool_kernel<<<(Nn * INNER + 255) / 256, 256, 0, stream>>>(xc, sg, pooled);

  bn_outproj_kernel<<<Ss / 16 / 4, 128, bn_lds, stream>>>(
      pooled, bn_gamma, bn_beta, bn_mean, bn_var, op_w, op_b, out);
}